// NonlinearMixtureMobile_69827578298464
// MI455X (gfx1250) — compile-verified
//
#include <hip/hip_runtime.h>
#include <hip/hip_bf16.h>

// ---------------------------------------------------------------------------
// MoE MobileNetV2 (top-1 routing) for MI455X / gfx1250.
// Only the argmax expert per image is evaluated (exact, 8x work saving).
// 1x1 convs -> per-image f32 WMMA GEMMs (v_wmma_f32_16x16x4_f32, exact f32),
// with the activation (B) operand staged in LDS, transposed + padded so each
// B fragment is one conflict-free ds_load_b64.
// ---------------------------------------------------------------------------

typedef __attribute__((ext_vector_type(2))) float v2f;
typedef __attribute__((ext_vector_type(8))) float v8f;

#define NB 64   // batch
#define NE 8    // experts

// ---------------------------------------------------------------------------
// Router: logits[b,e] = sum_{c,i,j} x * rw[e,c,i%4,j%4] + 64*rb[e]
// ---------------------------------------------------------------------------
__global__ __launch_bounds__(64)
void router_kernel(const float* __restrict__ x, const float* __restrict__ rw,
                   const float* __restrict__ rb,
                   int* __restrict__ idx, float* __restrict__ gate,
                   float* __restrict__ select0, float* __restrict__ loss_out,
                   float* __restrict__ ent_out)
{
    __shared__ float w[NE * 3 * 16];
    __shared__ float s_oh[NE], s_sel[NE], s_ent;
    const int t = threadIdx.x;           // one thread per image
    for (int i = t; i < NE * 48; i += 64) w[i] = rw[i];
    if (t < NE) { s_oh[t] = 0.f; s_sel[t] = 0.f; }
    if (t == 0) s_ent = 0.f;
    __syncthreads();

    float lg[NE];
    for (int e = 0; e < NE; ++e) lg[e] = 64.0f * rb[e];
    const float* xb = x + (long)t * 3 * 1024;
    for (int c = 0; c < 3; ++c)
        for (int i = 0; i < 32; ++i)
            for (int j = 0; j < 32; ++j) {
                float v = xb[c * 1024 + i * 32 + j];
                int wi = c * 16 + (i & 3) * 4 + (j & 3);
                #pragma unroll
                for (int e = 0; e < NE; ++e) lg[e] += v * w[e * 48 + wi];
            }
    const float invT = 1.0f / 1.5f;
    float mx = -1e30f;
    for (int e = 0; e < NE; ++e) { lg[e] *= invT; mx = fmaxf(mx, lg[e]); }
    float sum = 0.f, sel[NE];
    for (int e = 0; e < NE; ++e) { sel[e] = expf(lg[e] - mx); sum += sel[e]; }
    float inv = 1.0f / sum;
    int am = 0; float g = -1.f;
    for (int e = 0; e < NE; ++e) {
        sel[e] *= inv;
        if (sel[e] > g) { g = sel[e]; am = e; }
    }
    float ent = 0.f;
    for (int e = 0; e < NE; ++e) {
        ent -= sel[e] * logf(sel[e] + 1e-8f);
        select0[t * NE + e] = (e == am) ? 1.0f : 0.0f;
        atomicAdd(&s_sel[e], sel[e]);
    }
    atomicAdd(&s_oh[am], 1.0f);
    atomicAdd(&s_ent, ent);
    idx[t] = am;
    gate[t] = g;
    __syncthreads();
    if (t == 0) {
        float L = 0.f;
        for (int e = 0; e < NE; ++e) L += (s_sel[e] / 64.f) * (s_oh[e] / 64.f);
        *loss_out = (float)NE * L;      // mean_e(proxy*density)*E^2
        *ent_out = s_ent / 64.f;
    }
}

// ---------------------------------------------------------------------------
// conv1: 3x3, 3->32, pad 1, stride 1, 32x32, BN+ReLU.  Output [(b*32+c)*1024+p]
// ---------------------------------------------------------------------------
__global__ __launch_bounds__(256)
void conv1_kernel(const float* __restrict__ x, const float* __restrict__ W,
                  const float* __restrict__ G, const float* __restrict__ Bt,
                  float* __restrict__ Y, const int* __restrict__ idx)
{
    __shared__ float w[32 * 27];
    __shared__ float g[32], bb[32];
    const int b = blockIdx.x, t = threadIdx.x, e = idx[b];
    for (int i = t; i < 864; i += 256) w[i] = W[(long)e * 864 + i];
    if (t < 32) { g[t] = G[e * 32 + t]; bb[t] = Bt[e * 32 + t]; }
    __syncthreads();
    const float* xb = x + (long)b * 3 * 1024;
    for (int i = t; i < 32 * 1024; i += 256) {
        int c = i >> 10, p = i & 1023, y = p >> 5, xp = p & 31;
        float acc = 0.f;
        for (int ci = 0; ci < 3; ++ci)
            for (int ki = 0; ki < 3; ++ki) {
                int yi = y - 1 + ki; if (yi < 0 || yi > 31) continue;
                for (int kj = 0; kj < 3; ++kj) {
                    int xi = xp - 1 + kj; if (xi < 0 || xi > 31) continue;
                    acc += xb[ci * 1024 + yi * 32 + xi] * w[c * 27 + ci * 9 + ki * 3 + kj];
                }
            }
        float v = acc * g[c] + bb[c];
        Y[((long)b * 32 + c) * 1024 + p] = fmaxf(v, 0.f);
    }
}

// ---------------------------------------------------------------------------
// 1x1 conv as per-image WMMA GEMM:  Y[Cout,P] = act(BN(W[Cout,Cin] @ X[Cin,P]) [+res])
// Workgroup = one 16-wide N tile, 8 waves stacked in M (128 channels).
// X[:,ntile] staged in LDS transposed as xs[col][k] with stride Cin+2 so the
// B fragment is one aligned ds_load_b64; A fragment is one global float2.
// K stepped by 8 = 2x V_WMMA_F32_16X16X4_F32 per iteration (Cin % 8 == 0).
// ---------------------------------------------------------------------------
__global__ __launch_bounds__(256)
void gemm1x1_kernel(const float* __restrict__ X, const float* __restrict__ W,
                    const float* __restrict__ G, const float* __restrict__ Bt,
                    const float* __restrict__ R, float* __restrict__ Y,
                    const int* __restrict__ idx,
                    int Cin, int Cout, int P, int relu)
{
    extern __shared__ float xs[];              // 16 * (Cin+2) floats
    const int b = blockIdx.x;
    const int e = idx[b];
    const int n0 = blockIdx.y << 4;            // N tile base (P % 16 == 0)
    const int LD = Cin + 2;                    // padded LDS stride (even)
    const long Xb = (long)b * Cin * P;

    // cooperative transpose-stage of X[:, n0:n0+16] into LDS
    for (int i = threadIdx.x; i < Cin * 16; i += 256) {
        int k = i >> 4, c = i & 15;            // 16 consecutive cols -> coalesced
        xs[c * LD + k] = X[Xb + (long)k * P + n0 + c];
    }
    __syncthreads();

    const int wave = threadIdx.x >> 5;
    const int Mtiles = (Cout + 15) >> 4;
    const int mt = blockIdx.z * 8 + wave;
    if (mt >= Mtiles) return;                  // whole-wave exit: EXEC stays all-1s
    const int lane = threadIdx.x & 31;
    const int hi = lane >> 4;                  // half-wave: K pair select / M+8
    const int l  = lane & 15;
    const int m0 = mt << 4;
    const int mA = m0 + l;
    const int mc = (mA < Cout) ? mA : (Cout - 1);
    const float wmask = (mA < Cout) ? 1.0f : 0.0f;
    const float* Wrow = W + (long)e * Cout * Cin + (long)mc * Cin;
    const float* xrow = xs + l * LD;
    const int p = n0 + l;
    __builtin_prefetch(Wrow, 0, 3);            // global_prefetch_b8 on gfx1250

    v8f acc = {};
    for (int k0 = 0; k0 < Cin; k0 += 8) {
        v2f a0 = *(const v2f*)(Wrow + k0 + 2 * hi);          // global_load_b64
        v2f b0 = *(const v2f*)(xrow + k0 + 2 * hi);          // ds_load_b64
        acc = __builtin_amdgcn_wmma_f32_16x16x4_f32(
            false, a0 * wmask, false, b0, (short)0, acc, false, false);
        v2f a1 = *(const v2f*)(Wrow + k0 + 4 + 2 * hi);
        v2f b1 = *(const v2f*)(xrow + k0 + 4 + 2 * hi);
        acc = __builtin_amdgcn_wmma_f32_16x16x4_f32(
            false, a1 * wmask, false, b1, (short)0, acc, false, false);
    }

    // epilogue: BN affine, optional residual, optional ReLU
    #pragma unroll
    for (int r = 0; r < 8; ++r) {
        int m = m0 + r + hi * 8;
        if (m < Cout) {
            float v = acc[r] * G[e * Cout + m] + Bt[e * Cout + m];
            long o = ((long)b * Cout + m) * P + p;
            if (R) v += R[o];
            if (relu) v = fmaxf(v, 0.f);
            Y[o] = v;
        }
    }
}

// ---------------------------------------------------------------------------
// Depthwise 3x3 (pad 1, stride 1 or 2) + BN + ReLU
// ---------------------------------------------------------------------------
__global__ __launch_bounds__(256)
void dwconv_kernel(const float* __restrict__ H, const float* __restrict__ W2,
                   const float* __restrict__ G, const float* __restrict__ Bt,
                   float* __restrict__ D, const int* __restrict__ idx,
                   int C, int Sin, int Sout, int stride)
{
    const int b = blockIdx.x;
    const int i = blockIdx.y * 256 + threadIdx.x;
    const int Pout = Sout * Sout, Pin = Sin * Sin;
    if (i >= C * Pout) return;
    const int c = i / Pout, q = i - c * Pout;
    const int yo = q / Sout, xo = q - yo * Sout;
    const int e = idx[b];
    const float* w = W2 + ((long)e * C + c) * 9;
    const float* h = H + ((long)b * C + c) * Pin;
    float acc = 0.f;
    #pragma unroll
    for (int ki = 0; ki < 3; ++ki) {
        int yi = yo * stride - 1 + ki;
        if (yi < 0 || yi >= Sin) continue;
        #pragma unroll
        for (int kj = 0; kj < 3; ++kj) {
            int xi = xo * stride - 1 + kj;
            if (xi < 0 || xi >= Sin) continue;
            acc += h[yi * Sin + xi] * w[ki * 3 + kj];
        }
    }
    float v = acc * G[e * C + c] + Bt[e * C + c];
    D[((long)b * C + c) * Pout + q] = fmaxf(v, 0.f);
}

// ---------------------------------------------------------------------------
// Head: global mean over 4x4, linear 1280->10, scale by gate, softmax.
// ---------------------------------------------------------------------------
__global__ __launch_bounds__(256)
void head_kernel(const float* __restrict__ A2, const float* __restrict__ lw,
                 const float* __restrict__ lb, const int* __restrict__ idx,
                 const float* __restrict__ gate, float* __restrict__ out)
{
    __shared__ float mean[1280];
    __shared__ float red[256];
    __shared__ float lg[10];
    const int b = blockIdx.x, t = threadIdx.x, e = idx[b];
    for (int c = t; c < 1280; c += 256) {
        const float* p = A2 + ((long)b * 1280 + c) * 16;
        float s = 0.f;
        #pragma unroll
        for (int i = 0; i < 16; ++i) s += p[i];
        mean[c] = s * (1.0f / 16.0f);
    }
    __syncthreads();
    for (int k = 0; k < 10; ++k) {
        const float* w = lw + ((long)e * 10 + k) * 1280;
        float s = 0.f;
        for (int c = t; c < 1280; c += 256) s += w[c] * mean[c];
        red[t] = s; __syncthreads();
        for (int st = 128; st > 0; st >>= 1) {
            if (t < st) red[t] += red[t + st];
            __syncthreads();
        }
        if (t == 0) lg[k] = red[0] + lb[e * 10 + k];
        __syncthreads();
    }
    if (t == 0) {
        float g = gate[b], mx = -1e30f;
        float v[10];
        for (int k = 0; k < 10; ++k) { v[k] = lg[k] * g; mx = fmaxf(mx, v[k]); }
        float s = 0.f;
        for (int k = 0; k < 10; ++k) { v[k] = expf(v[k] - mx); s += v[k]; }
        float inv = 1.0f / s;
        for (int k = 0; k < 10; ++k) out[b * 10 + k] = v[k] * inv;
    }
}

// ---------------------------------------------------------------------------
// Host-side orchestration
// ---------------------------------------------------------------------------
extern "C" void kernel_launch(void* const* d_in, const int* in_sizes, int n_in,
                              void* d_out, int out_size, void* d_ws, size_t ws_size,
                              hipStream_t stream)
{
    struct Meta { int hid, inp, outp, st, sc; };
    static const Meta M[17] = {
        { 32,  32,  16, 1, 1},
        { 96,  16,  24, 1, 1}, {144,  24,  24, 1, 0},
        {144,  24,  32, 2, 0}, {192,  32,  32, 1, 0}, {192,  32,  32, 1, 0},
        {192,  32,  64, 2, 0}, {384,  64,  64, 1, 0}, {384,  64,  64, 1, 0}, {384,  64,  64, 1, 0},
        {384,  64,  96, 1, 1}, {576,  96,  96, 1, 0}, {576,  96,  96, 1, 0},
        {576,  96, 160, 2, 0}, {960, 160, 160, 1, 0}, {960, 160, 160, 1, 0},
        {960, 160, 320, 1, 1}
    };

    // --- unpack inputs (setup_inputs dict order, params flattened depth-first) ---
    const float* x  = (const float*)d_in[0];
    const float* rw = (const float*)d_in[1];
    const float* rb = (const float*)d_in[2];
    int pi = 3;
    const float* conv1w = (const float*)d_in[pi++];
    const float* bn1g   = (const float*)d_in[pi++];
    const float* bn1b   = (const float*)d_in[pi++];
    const float* bw[17][12];
    for (int i = 0; i < 17; ++i) {
        for (int j = 0; j < 9; ++j) bw[i][j] = (const float*)d_in[pi++];
        if (M[i].sc) { for (int j = 9; j < 12; ++j) bw[i][j] = (const float*)d_in[pi++]; }
        else         { bw[i][9] = bw[i][10] = bw[i][11] = nullptr; }
    }
    const float* conv2w = (const float*)d_in[pi++];
    const float* bn2g   = (const float*)d_in[pi++];
    const float* bn2b   = (const float*)d_in[pi++];
    const float* linw   = (const float*)d_in[pi++];
    const float* linb   = (const float*)d_in[pi++];

    // --- outputs: (out[64,10], select0[64,8], loss, entropy) concat ---
    float* out = (float*)d_out;
    float* o_sel  = out + 640;
    float* o_loss = out + 1152;
    float* o_ent  = out + 1153;

    // --- workspace layout ---
    const long ACT = (long)NB * 32768;        // inter-layer acts (max C*P = 32768)
    const long HID = (long)NB * 147456;       // max hid*P = 144*1024
    float* f      = (float*)d_ws;
    int*   d_idx  = (int*)f;                  // 64 ints
    float* d_gate = f + 64;                   // 64 floats
    float* bufA   = f + 128;
    float* bufB   = bufA + ACT;
    float* bufR   = bufB + ACT;
    float* bufH   = bufR + ACT;
    float* bufD   = bufH + HID;
    (void)ws_size; (void)n_in; (void)in_sizes; (void)out_size;

    auto gemm = [&](const float* X, const float* Wp, const float* Gp, const float* Bp,
                    const float* Rp, float* Yp, int Cin, int Cout, int P, int relu) {
        int Mt = (Cout + 15) / 16, Nt = P / 16;
        int gz = (Mt + 7) / 8;
        size_t shmem = (size_t)16 * (Cin + 2) * sizeof(float);
        gemm1x1_kernel<<<dim3(NB, Nt, gz), dim3(256), shmem, stream>>>(
            X, Wp, Gp, Bp, Rp, Yp, d_idx, Cin, Cout, P, relu);
    };

    // 1) routing + stats
    router_kernel<<<1, 64, 0, stream>>>(x, rw, rb, d_idx, d_gate, o_sel, o_loss, o_ent);
    // 2) stem
    conv1_kernel<<<NB, 256, 0, stream>>>(x, conv1w, bn1g, bn1b, bufA, d_idx);

    // 3) 17 inverted-residual blocks
    float* cur = bufA;
    float* nxt = bufB;
    int S = 32;
    for (int i = 0; i < 17; ++i) {
        const int Sin = S, Sout = (M[i].st == 2) ? S / 2 : S;
        S = Sout;
        const int Pin = Sin * Sin, Pout = Sout * Sout;
        // expand 1x1 + BN + ReLU  (WMMA)
        gemm(cur, bw[i][0], bw[i][1], bw[i][2], nullptr, bufH, M[i].inp, M[i].hid, Pin, 1);
        // depthwise 3x3 + BN + ReLU
        int total = M[i].hid * Pout;
        dwconv_kernel<<<dim3(NB, (total + 255) / 256), 256, 0, stream>>>(
            bufH, bw[i][3], bw[i][4], bw[i][5], bufD, d_idx, M[i].hid, Sin, Sout, M[i].st);
        // residual source
        const float* res = nullptr;
        if (M[i].st == 1) {
            if (M[i].sc) {   // 1x1 shortcut conv + BN (WMMA)
                gemm(cur, bw[i][9], bw[i][10], bw[i][11], nullptr, bufR,
                     M[i].inp, M[i].outp, Pout, 0);
                res = bufR;
            } else {
                res = cur;   // identity shortcut
            }
        }
        // project 1x1 + BN (+residual, no ReLU)  (WMMA)
        gemm(bufD, bw[i][6], bw[i][7], bw[i][8], res, nxt, M[i].hid, M[i].outp, Pout, 0);
        float* t = cur; cur = nxt; nxt = t;
    }

    // 4) conv2 head 320->1280 @4x4 + BN + ReLU  (WMMA)
    gemm(cur, conv2w, bn2g, bn2b, nullptr, nxt, 320, 1280, 16, 1);
    // 5) avgpool + linear + gate-scale + softmax
    head_kernel<<<NB, 256, 0, stream>>>(nxt, linw, linb, d_idx, d_gate, out);
}